// ResGatedGNN_41979010351140
// MI455X (gfx1250) — compile-verified
//
#include <hip/hip_runtime.h>

// ---------------------------------------------------------------------------
// GatedGraphConv GNN for MI455X (gfx1250), fp32 end-to-end.
// All GEMMs use V_WMMA_F32_16X16X4_F32 (wave32 WMMA, full fp32 precision).
// lin_in (dominant GEMM): 32-row register blocking (12 f32 accumulators),
// 2x-unrolled double-buffered software pipeline with two NAMED fragment sets,
// and a peeled epilogue so every in-loop address is affine in k (pointer
// increments only -> no spills, no per-iteration loadcnt-0 drain).
// Edge aggregation: L2-resident global_atomic_add_f32 (m/agg = 19.2MB << 192MB L2).
// ---------------------------------------------------------------------------

typedef __attribute__((ext_vector_type(2))) float v2f;
typedef __attribute__((ext_vector_type(8))) float v8f;

#define N_NODES 50000
#define E_EDGES 800000
#define IN_C    2000
#define HID     96
#define OUT_C   16
#define N_LAYERS 3
#define M_TILES   (N_NODES / 16)        // 3125, exact
#define M_TILES32 ((N_NODES + 31) / 32) // 1563 (last tile ragged: 16 rows)

// ---- WMMA fragment helpers (layouts per CDNA5 ISA 7.12.2) ------

// A-matrix 16x4 f32: lanes 0-15 -> M=0..15 with K=k0,k0+1 in the 2 VGPRs;
// lanes 16-31 -> same M rows with K=k0+2,k0+3.
__device__ __forceinline__ v2f load_a_frag(const float* __restrict__ A, int lda,
                                           int row0, int k0, int lane) {
    int m  = lane & 15;
    int kh = (lane >> 4) << 1;            // 0 or 2
    const float* p = A + (size_t)(row0 + m) * lda + (k0 + kh);
    v2f a; a.x = p[0]; a.y = p[1];
    return a;
}

// B-matrix 4x16 f32 (row-major B[k][n]): lane half selects K pair, lane&15 = N.
__device__ __forceinline__ v2f load_b_frag(const float* __restrict__ B, int ldb,
                                           int k0, int n0, int lane) {
    int n  = lane & 15;
    int kh = (lane >> 4) << 1;
    const float* p = B + (size_t)(k0 + kh) * ldb + (n0 + n);
    v2f b; b.x = p[0]; b.y = p[ldb];
    return b;
}

// Transposed B: logical B[k][n] = W[n][k] (for gi = m @ w_ih^T style GEMMs).
__device__ __forceinline__ v2f load_bt_frag(const float* __restrict__ W, int ldw,
                                            int k0, int n0, int lane) {
    int n  = lane & 15;
    int kh = (lane >> 4) << 1;
    const float* p = W + (size_t)(n0 + n) * ldw + (k0 + kh);
    v2f b; b.x = p[0]; b.y = p[1];
    return b;
}

// C/D 16x16 f32: lane&15 = N, lane>=16 -> rows 8..15, VGPR r -> row offset.
__device__ __forceinline__ void store_c_frag(float* __restrict__ C, int ldc,
                                             int row0, int n0, int lane, v8f c) {
    int n     = lane & 15;
    int mbase = (lane >> 4) * 8;
#pragma unroll
    for (int r = 0; r < 8; ++r)
        C[(size_t)(row0 + mbase + r) * ldc + (n0 + n)] = c[r];
}

__device__ __forceinline__ v8f bcast_bias(const float* __restrict__ b, int n0, int lane) {
    float bv = b[n0 + (lane & 15)];
    v8f c;
#pragma unroll
    for (int r = 0; r < 8; ++r) c[r] = bv;
    return c;
}

#define WMMA_F32(a, b, c) \
    __builtin_amdgcn_wmma_f32_16x16x4_f32(false, (a), false, (b), (short)0, (c), false, false)

__device__ __forceinline__ float sigmoidf_(float x) {
    return 1.0f / (1.0f + __expf(-x));
}

// ---------------------------------------------------------------------------
// Kernel 1: h = x @ W_in + b_in       [50000x2000]*[2000x96]
// One wave computes a 32x96 output tile. k-loop unrolled 2x with two named
// fragment sets: phase 0 computes on setA while setB's loads fly, phase 1
// vice-versa. The last k-step (k=1992/1996) is PEELED so all in-loop load
// addresses are affine in k (no clamp select -> strength-reduced pointer
// increments, no spilled bases, no full loadcnt drain). Ragged tail tile
// clamps its second row pointer (EXEC all-ones through every WMMA) and
// skips the second store.
// ---------------------------------------------------------------------------
__global__ void lin_in_kernel(const float* __restrict__ x, const float* __restrict__ W,
                              const float* __restrict__ bias, float* __restrict__ h) {
    int lane = threadIdx.x & 31;
    int tile = blockIdx.x * (blockDim.x >> 5) + (threadIdx.x >> 5);
    if (tile >= M_TILES32) return;            // whole-wave exit; EXEC all-1 inside
    int row0 = tile * 32;
    bool full = (row0 + 32 <= N_NODES);
    int row1 = full ? (row0 + 16) : row0;     // tail: duplicate rows, store skipped

    // Per-lane A-stream base pointers (reused for the prefetch address).
    int m  = lane & 15;
    int kh = (lane >> 4) << 1;
    const float* pa0 = x + (size_t)(row0 + m) * IN_C + kh;
    const float* pa1 = x + (size_t)(row1 + m) * IN_C + kh;

    v8f acc0[6], acc1[6];
#pragma unroll
    for (int j = 0; j < 6; ++j) {
        acc0[j] = bcast_bias(bias, j * 16, lane);
        acc1[j] = acc0[j];
    }

    // ---- pipeline prologue: set A fragments at k = 0 ----
    v2f a0A = load_a_frag(x, IN_C, row0, 0, lane);
    v2f a1A = load_a_frag(x, IN_C, row1, 0, lane);
    v2f bA[6];
#pragma unroll
    for (int j = 0; j < 6; ++j) bA[j] = load_b_frag(W, HID, 0, j * 16, lane);

    v2f a0B, a1B, bB[6];

    // Steady state: k = 0, 8, ..., IN_C-16. All addresses affine in k.
#pragma unroll 1
    for (int k = 0; k + 8 < IN_C; k += 8) {
        // Unconditional streaming prefetch, +1KB ahead on the A base pointers.
        __builtin_prefetch(pa0 + k + 256, 0, 0);   // global_prefetch_b8
        __builtin_prefetch(pa1 + k + 256, 0, 0);

        // -- phase 0: issue set B loads (k+4), compute on set A (k) --
        a0B = load_a_frag(x, IN_C, row0, k + 4, lane);
        a1B = load_a_frag(x, IN_C, row1, k + 4, lane);
#pragma unroll
        for (int j = 0; j < 6; ++j) bB[j] = load_b_frag(W, HID, k + 4, j * 16, lane);
#pragma unroll
        for (int j = 0; j < 6; ++j) {
            acc0[j] = WMMA_F32(a0A, bA[j], acc0[j]);
            acc1[j] = WMMA_F32(a1A, bA[j], acc1[j]);
        }

        // -- phase 1: issue set A loads (k+8), compute on set B (k+4) --
        a0A = load_a_frag(x, IN_C, row0, k + 8, lane);
        a1A = load_a_frag(x, IN_C, row1, k + 8, lane);
#pragma unroll
        for (int j = 0; j < 6; ++j) bA[j] = load_b_frag(W, HID, k + 8, j * 16, lane);
#pragma unroll
        for (int j = 0; j < 6; ++j) {
            acc0[j] = WMMA_F32(a0B, bB[j], acc0[j]);
            acc1[j] = WMMA_F32(a1B, bB[j], acc1[j]);
        }
    }

    // ---- peeled epilogue: k = IN_C-8 (setA already loaded by the loop) ----
    {
        const int kl = IN_C - 4;                 // 1996
        a0B = load_a_frag(x, IN_C, row0, kl, lane);
        a1B = load_a_frag(x, IN_C, row1, kl, lane);
#pragma unroll
        for (int j = 0; j < 6; ++j) bB[j] = load_b_frag(W, HID, kl, j * 16, lane);
#pragma unroll
        for (int j = 0; j < 6; ++j) {
            acc0[j] = WMMA_F32(a0A, bA[j], acc0[j]);
            acc1[j] = WMMA_F32(a1A, bA[j], acc1[j]);
        }
#pragma unroll
        for (int j = 0; j < 6; ++j) {
            acc0[j] = WMMA_F32(a0B, bB[j], acc0[j]);
            acc1[j] = WMMA_F32(a1B, bB[j], acc1[j]);
        }
    }

#pragma unroll
    for (int j = 0; j < 6; ++j) store_c_frag(h, HID, row0, j * 16, lane, acc0[j]);
    if (full) {
#pragma unroll
        for (int j = 0; j < 6; ++j) store_c_frag(h, HID, row1, j * 16, lane, acc1[j]);
    }
}

// ---------------------------------------------------------------------------
// Kernel 2: m = h @ Wc                [50000x96]*[96x96], same M=32 blocking.
// K=96 -> 24 steps, fully unrolled so the scheduler hoists loads across steps.
// ---------------------------------------------------------------------------
__global__ void conv_kernel(const float* __restrict__ h, const float* __restrict__ Wc,
                            float* __restrict__ m) {
    int lane = threadIdx.x & 31;
    int tile = blockIdx.x * (blockDim.x >> 5) + (threadIdx.x >> 5);
    if (tile >= M_TILES32) return;
    int row0 = tile * 32;
    bool full = (row0 + 32 <= N_NODES);
    int row1 = full ? (row0 + 16) : row0;

    v8f acc0[6], acc1[6];
#pragma unroll
    for (int j = 0; j < 6; ++j) {
        v8f z;
#pragma unroll
        for (int r = 0; r < 8; ++r) z[r] = 0.0f;
        acc0[j] = z;
        acc1[j] = z;
    }
#pragma unroll
    for (int k = 0; k < HID; k += 4) {
        v2f a0 = load_a_frag(h, HID, row0, k, lane);
        v2f a1 = load_a_frag(h, HID, row1, k, lane);
        v2f bf[6];
#pragma unroll
        for (int j = 0; j < 6; ++j) bf[j] = load_b_frag(Wc, HID, k, j * 16, lane);
#pragma unroll
        for (int j = 0; j < 6; ++j) {
            acc0[j] = WMMA_F32(a0, bf[j], acc0[j]);
            acc1[j] = WMMA_F32(a1, bf[j], acc1[j]);
        }
    }
#pragma unroll
    for (int j = 0; j < 6; ++j) store_c_frag(m, HID, row0, j * 16, lane, acc0[j]);
    if (full) {
#pragma unroll
        for (int j = 0; j < 6; ++j) store_c_frag(m, HID, row1, j * 16, lane, acc1[j]);
    }
}

// ---------------------------------------------------------------------------
// Kernel 3: zero the aggregation buffer
// ---------------------------------------------------------------------------
__global__ void zero_kernel(float* __restrict__ p, int n) {
    int i = blockIdx.x * blockDim.x + threadIdx.x;
    if (i < n) p[i] = 0.0f;
}

// ---------------------------------------------------------------------------
// Kernel 4: agg[dst[e]] += m[src[e]]  (scatter-add over 800000 edges x 96)
// edge_index is int64 in the reference. m/agg are L2-resident -> fast atomics.
// ---------------------------------------------------------------------------
__global__ void edge_scatter_kernel(const long long* __restrict__ src,
                                    const long long* __restrict__ dst,
                                    const float* __restrict__ m,
                                    float* __restrict__ agg, int nwork) {
    int t = blockIdx.x * blockDim.x + threadIdx.x;
    if (t >= nwork) return;
    int e = t / HID;
    int c = t - e * HID;
    int s = (int)src[e];
    int d = (int)dst[e];
    float v = m[(size_t)s * HID + c];
    unsafeAtomicAdd(&agg[(size_t)d * HID + c], v);   // global_atomic_add_f32
}

// ---------------------------------------------------------------------------
// Kernel 5: fused GRU cell  h_new = GRUCell(agg, h)
// Per 16-row tile, per 16-col slice j: 6 WMMA accumulators =
//   i_r,i_z,i_n from agg @ w_ih^T  (cols j, 96+j, 192+j)
//   h_r,h_z,h_n from h   @ w_hh^T
// Gate math stays in-register; gate matrices never touch memory.
// ---------------------------------------------------------------------------
__global__ void gru_kernel(const float* __restrict__ agg, const float* __restrict__ h,
                           const float* __restrict__ w_ih, const float* __restrict__ w_hh,
                           const float* __restrict__ b_ih, const float* __restrict__ b_hh,
                           float* __restrict__ h_new) {
    int lane = threadIdx.x & 31;
    int tile = blockIdx.x * (blockDim.x >> 5) + (threadIdx.x >> 5);
    if (tile >= M_TILES) return;
    int row0  = tile * 16;
    int n     = lane & 15;
    int mbase = (lane >> 4) * 8;

#pragma unroll 1
    for (int j = 0; j < 6; ++j) {
        int cr = j * 16;            // reset-gate columns
        int cz = HID + j * 16;      // update-gate columns
        int cn = 2 * HID + j * 16;  // new-gate columns

        v8f ir = bcast_bias(b_ih, cr, lane);
        v8f iz = bcast_bias(b_ih, cz, lane);
        v8f inn = bcast_bias(b_ih, cn, lane);
        v8f hr = bcast_bias(b_hh, cr, lane);
        v8f hz = bcast_bias(b_hh, cz, lane);
        v8f hn = bcast_bias(b_hh, cn, lane);

#pragma unroll
        for (int k = 0; k < HID; k += 4) {
            // ---- load phase ----
            v2f aA = load_a_frag(agg, HID, row0, k, lane);
            v2f aH = load_a_frag(h,   HID, row0, k, lane);
            v2f bir = load_bt_frag(w_ih, HID, k, cr, lane);
            v2f biz = load_bt_frag(w_ih, HID, k, cz, lane);
            v2f bin = load_bt_frag(w_ih, HID, k, cn, lane);
            v2f bhr = load_bt_frag(w_hh, HID, k, cr, lane);
            v2f bhz = load_bt_frag(w_hh, HID, k, cz, lane);
            v2f bhn = load_bt_frag(w_hh, HID, k, cn, lane);
            // ---- compute phase: 6-WMMA burst ----
            ir  = WMMA_F32(aA, bir, ir);
            iz  = WMMA_F32(aA, biz, iz);
            inn = WMMA_F32(aA, bin, inn);
            hr  = WMMA_F32(aH, bhr, hr);
            hz  = WMMA_F32(aH, bhz, hz);
            hn  = WMMA_F32(aH, bhn, hn);
        }

#pragma unroll
        for (int r = 0; r < 8; ++r) {
            float rg = sigmoidf_(ir[r] + hr[r]);
            float zg = sigmoidf_(iz[r] + hz[r]);
            float ng = tanhf(inn[r] + rg * hn[r]);
            size_t idx = (size_t)(row0 + mbase + r) * HID + (cr + n);
            float hv = h[idx];
            h_new[idx] = (1.0f - zg) * ng + zg * hv;
        }
    }
}

// ---------------------------------------------------------------------------
// Kernel 6: out = h @ W_out + b_out   [50000x96]*[96x16]
// ---------------------------------------------------------------------------
__global__ void lin_out_kernel(const float* __restrict__ h, const float* __restrict__ W,
                               const float* __restrict__ bias, float* __restrict__ out) {
    int lane = threadIdx.x & 31;
    int tile = blockIdx.x * (blockDim.x >> 5) + (threadIdx.x >> 5);
    if (tile >= M_TILES) return;
    int row0 = tile * 16;

    v8f acc = bcast_bias(bias, 0, lane);
#pragma unroll
    for (int k = 0; k < HID; k += 4) {
        v2f a = load_a_frag(h, HID, row0, k, lane);
        v2f b = load_b_frag(W, OUT_C, k, 0, lane);
        acc = WMMA_F32(a, b, acc);
    }
    store_c_frag(out, OUT_C, row0, 0, lane, acc);
}

// ---------------------------------------------------------------------------
// Launch: lin_in -> 3x { conv, zero, scatter, gru } -> lin_out
// h ping-pongs between two workspace buffers (gru reads h, writes h_next).
// ---------------------------------------------------------------------------
extern "C" void kernel_launch(void* const* d_in, const int* in_sizes, int n_in,
                              void* d_out, int out_size, void* d_ws, size_t ws_size,
                              hipStream_t stream) {
    const float*     x        = (const float*)d_in[0];
    const long long* ei       = (const long long*)d_in[1];   // int64 [2, E]
    const float*     W_in     = (const float*)d_in[2];
    const float*     b_in     = (const float*)d_in[3];
    const float*     conv_w   = (const float*)d_in[4];       // [L, H, H]
    const float*     gru_w_ih = (const float*)d_in[5];       // [3H, H]
    const float*     gru_w_hh = (const float*)d_in[6];
    const float*     gru_b_ih = (const float*)d_in[7];
    const float*     gru_b_hh = (const float*)d_in[8];
    const float*     W_out    = (const float*)d_in[9];
    const float*     b_out    = (const float*)d_in[10];
    float*           out      = (float*)d_out;

    const size_t NH = (size_t)N_NODES * HID;
    float* hA   = (float*)d_ws;
    float* hB   = hA + NH;
    float* mbuf = hB + NH;
    float* agg  = mbuf + NH;

    const long long* src = ei;              // row 0
    const long long* dst = ei + E_EDGES;    // row 1

    dim3 blk(128);                               // 4 waves/block
    dim3 grd16((M_TILES + 3) / 4);               // 16-row wave-tiles
    dim3 grd32((M_TILES32 + 3) / 4);             // 32-row wave-tiles

    lin_in_kernel<<<grd32, blk, 0, stream>>>(x, W_in, b_in, hA);

    const int nh_elems = N_NODES * HID;          // 4.8M
    const int nwork    = E_EDGES * HID;          // 76.8M

    float* hcur = hA;
    float* hnext = hB;
    for (int l = 0; l < N_LAYERS; ++l) {
        conv_kernel<<<grd32, blk, 0, stream>>>(
            hcur, conv_w + (size_t)l * HID * HID, mbuf);
        zero_kernel<<<(nh_elems + 255) / 256, 256, 0, stream>>>(agg, nh_elems);
        edge_scatter_kernel<<<(nwork + 255) / 256, 256, 0, stream>>>(
            src, dst, mbuf, agg, nwork);
        gru_kernel<<<grd16, blk, 0, stream>>>(
            agg, hcur, gru_w_ih, gru_w_hh, gru_b_ih, gru_b_hh, hnext);
        float* t = hcur; hcur = hnext; hnext = t;
    }

    lin_out_kernel<<<grd16, blk, 0, stream>>>(hcur, W_out, b_out, out);
}